// Net_85753317032178
// MI455X (gfx1250) — compile-verified
//
#include <hip/hip_runtime.h>
#include <math.h>

typedef float v2f __attribute__((ext_vector_type(2)));
typedef float v8f __attribute__((ext_vector_type(8)));

// Branchless ELU: exp taken only on the x<=0 side where |result|<1, so the
// hardware v_exp_f32 approximation is ample; fminf keeps the unconditional
// exp from overflowing for large positive x.
__device__ __forceinline__ float elu1(float x) {
    const float e = __expf(fminf(x, 0.0f)) - 1.0f;
    return x > 0.0f ? x : e;
}

// ---------------------------------------------------------------------------
// Encoder: h = elu(elu(x @ w1 + b1) @ w2 + b2)   x:(128x16) per graph
// One workgroup per graph, 8 waves, WMMA f32 16x16x4 tiles.
// ---------------------------------------------------------------------------
__global__ __launch_bounds__(256) void k_encoder(
    const float* __restrict__ x,
    const float* __restrict__ w1, const float* __restrict__ b1,
    const float* __restrict__ w2, const float* __restrict__ b2,
    float* __restrict__ hout)
{
    __shared__ float xs[128 * 17];     // padded stride 17 (gcd(17,64)=1)
    __shared__ float h1[128 * 130];    // padded stride 130
    const int g = blockIdx.x, tid = threadIdx.x;
    const float* xg = x + g * 128 * 16;
    for (int t = tid; t < 128 * 16; t += 256)
        xs[(t >> 4) * 17 + (t & 15)] = xg[t];
    __syncthreads();

    const int wv = tid >> 5, lane = tid & 31, hf = lane >> 4, l = lane & 15;

    // layer 1: (128x16)@(16x128)
    for (int t = wv; t < 64; t += 8) {
        const int m0 = (t >> 3) << 4, n0 = (t & 7) << 4;
        v8f c = {};
        for (int k = 0; k < 16; k += 4) {
            const int ka = k + hf * 2;
            v2f a, b;
            a.x = xs[(m0 + l) * 17 + ka];
            a.y = xs[(m0 + l) * 17 + ka + 1];
            b.x = w1[ka * 128 + n0 + l];
            b.y = w1[(ka + 1) * 128 + n0 + l];
            c = __builtin_amdgcn_wmma_f32_16x16x4_f32(false, a, false, b,
                                                      (short)0, c, false, false);
        }
#pragma unroll
        for (int r = 0; r < 8; ++r) {
            const int row = m0 + hf * 8 + r, col = n0 + l;
            h1[row * 130 + col] = elu1(c[r] + b1[col]);
        }
    }
    __syncthreads();

    // layer 2: (128x128)@(128x128)
    float* ho = hout + g * 128 * 128;
    for (int t = wv; t < 64; t += 8) {
        const int m0 = (t >> 3) << 4, n0 = (t & 7) << 4;
        v8f c = {};
        for (int k = 0; k < 128; k += 4) {
            const int ka = k + hf * 2;
            v2f a, b;
            a.x = h1[(m0 + l) * 130 + ka];
            a.y = h1[(m0 + l) * 130 + ka + 1];
            b.x = w2[ka * 128 + n0 + l];
            b.y = w2[(ka + 1) * 128 + n0 + l];
            c = __builtin_amdgcn_wmma_f32_16x16x4_f32(false, a, false, b,
                                                      (short)0, c, false, false);
        }
#pragma unroll
        for (int r = 0; r < 8; ++r) {
            const int row = m0 + hf * 8 + r, col = n0 + l;
            ho[row * 128 + col] = elu1(c[r] + b2[col]);
        }
    }
}

// ---------------------------------------------------------------------------
// kNN: d2 = sq_i + sq_j - 2 * (h h^T)  via WMMA Gram; 24-pass min-select.
// ---------------------------------------------------------------------------
__global__ __launch_bounds__(256) void k_knn(
    const float* __restrict__ h, int* __restrict__ idxg)
{
    __shared__ float hs[128 * 130];
    __shared__ float d2[128 * 129];   // padded stride 129 for row scans
    __shared__ float sq[128];
    const int g = blockIdx.x, tid = threadIdx.x;
    const float* hg = h + g * 16384;
    for (int t = tid; t < 16384; t += 256)
        hs[(t >> 7) * 130 + (t & 127)] = hg[t];
    __syncthreads();
    if (tid < 128) {
        float s = 0.0f;
        for (int k = 0; k < 128; ++k) { const float v = hs[tid * 130 + k]; s += v * v; }
        sq[tid] = s;
    }
    __syncthreads();

    const int wv = tid >> 5, lane = tid & 31, hf = lane >> 4, l = lane & 15;
    for (int t = wv; t < 64; t += 8) {
        const int m0 = (t >> 3) << 4, n0 = (t & 7) << 4;
        v8f c = {};
        for (int k = 0; k < 128; k += 4) {
            const int ka = k + hf * 2;
            v2f a, b;
            a.x = hs[(m0 + l) * 130 + ka];
            a.y = hs[(m0 + l) * 130 + ka + 1];
            b.x = hs[(n0 + l) * 130 + ka];      // B = h^T (transposed read)
            b.y = hs[(n0 + l) * 130 + ka + 1];
            c = __builtin_amdgcn_wmma_f32_16x16x4_f32(false, a, false, b,
                                                      (short)0, c, false, false);
        }
#pragma unroll
        for (int r = 0; r < 8; ++r) {
            const int row = m0 + hf * 8 + r, col = n0 + l;
            d2[row * 129 + col] = sq[row] + sq[col] - 2.0f * c[r];
        }
    }
    __syncthreads();

    if (tid < 128) {
        int* orow = idxg + g * 128 * 24 + tid * 24;
        float* drow = d2 + tid * 129;
        for (int p = 0; p < 24; ++p) {
            float best = 3.4e38f; int bj = 0;
            for (int j = 0; j < 128; ++j) {
                const float v = drow[j];
                if (v < best) { best = v; bj = j; }
            }
            orow[p] = bj;
            drow[bj] = 3.4e38f;
        }
    }
}

// ---------------------------------------------------------------------------
// Edge-MLP factorization: A = h@(Wt - Wb) + b,  Bm = h@Wb     (per graph)
// ---------------------------------------------------------------------------
__global__ __launch_bounds__(256) void k_convAB(
    const float* __restrict__ h,
    const float* __restrict__ w, const float* __restrict__ bias,
    float* __restrict__ Aout, float* __restrict__ Bout)
{
    __shared__ float hs[128 * 130];
    const int g = blockIdx.x, tid = threadIdx.x;
    const float* hg = h + g * 16384;
    for (int t = tid; t < 16384; t += 256)
        hs[(t >> 7) * 130 + (t & 127)] = hg[t];
    __syncthreads();

    const int wv = tid >> 5, lane = tid & 31, hf = lane >> 4, l = lane & 15;
    const float* wt = w;                 // rows 0..127   (x_i part)
    const float* wb = w + 128 * 128;     // rows 128..255 (x_j - x_i part)

    for (int t = wv; t < 128; t += 8) {
        const bool isA = (t < 64);       // uniform per wave
        const int tt = isA ? t : t - 64;
        const int m0 = (tt >> 3) << 4, n0 = (tt & 7) << 4;
        v8f c = {};
        for (int k = 0; k < 128; k += 4) {
            const int ka = k + hf * 2;
            v2f a, b;
            a.x = hs[(m0 + l) * 130 + ka];
            a.y = hs[(m0 + l) * 130 + ka + 1];
            if (isA) {
                b.x = wt[ka * 128 + n0 + l] - wb[ka * 128 + n0 + l];
                b.y = wt[(ka + 1) * 128 + n0 + l] - wb[(ka + 1) * 128 + n0 + l];
            } else {
                b.x = wb[ka * 128 + n0 + l];
                b.y = wb[(ka + 1) * 128 + n0 + l];
            }
            c = __builtin_amdgcn_wmma_f32_16x16x4_f32(false, a, false, b,
                                                      (short)0, c, false, false);
        }
        float* out = (isA ? Aout : Bout) + g * 16384;
#pragma unroll
        for (int r = 0; r < 8; ++r) {
            const int row = m0 + hf * 8 + r, col = n0 + l;
            out[row * 128 + col] = c[r] + (isA ? bias[col] : 0.0f);
        }
    }
}

// ---------------------------------------------------------------------------
// Edge combine + maxpool: h_i = elu(A_i + max_{j in knn(i)} Bm_j)
// (ELU is monotone, so max commutes through it.)  One wave per point.
// ---------------------------------------------------------------------------
__global__ __launch_bounds__(256) void k_edge(
    const float* __restrict__ Ain, const float* __restrict__ Bin,
    const int* __restrict__ idxg, float* __restrict__ hout)
{
    __shared__ float4 Bs[128 * 32];  // 64 KB
    const int g = blockIdx.x, tid = threadIdx.x;
    const float4* Bg = (const float4*)(Bin + g * 16384);
    for (int t = tid; t < 4096; t += 256) Bs[t] = Bg[t];
    __syncthreads();

    const int wv = tid >> 5, lane = tid & 31;
    const float4* Ag = (const float4*)(Ain + g * 16384);
    float4* Hg = (float4*)(hout + g * 16384);
    const int* ig = idxg + g * 128 * 24;

    for (int i = wv; i < 128; i += 8) {
        const float4 a = Ag[i * 32 + lane];
        float mx = -3.4e38f, my = -3.4e38f, mz = -3.4e38f, mw = -3.4e38f;
        for (int p = 0; p < 24; ++p) {
            const int j = ig[i * 24 + p];           // uniform across wave
            const float4 b = Bs[j * 32 + lane];
            mx = fmaxf(mx, b.x); my = fmaxf(my, b.y);
            mz = fmaxf(mz, b.z); mw = fmaxf(mw, b.w);
        }
        float4 o;
        o.x = elu1(a.x + mx); o.y = elu1(a.y + my);
        o.z = elu1(a.z + mz); o.w = elu1(a.w + mw);
        Hg[i * 32 + lane] = o;
    }
}

// ---------------------------------------------------------------------------
// Sum-pool over points per graph.
// ---------------------------------------------------------------------------
__global__ __launch_bounds__(128) void k_pool(
    const float* __restrict__ h, float* __restrict__ pooled)
{
    const int g = blockIdx.x, c = threadIdx.x;
    const float* hg = h + g * 16384;
    float s = 0.0f;
    for (int i = 0; i < 128; ++i) s += hg[i * 128 + c];
    pooled[g * 128 + c] = s;
}

// ---------------------------------------------------------------------------
// Output MLP: 128 -> 64 -> 32 -> 32 -> 6 (elu except last). One block/graph.
// ---------------------------------------------------------------------------
__global__ __launch_bounds__(64) void k_mlp(
    const float* __restrict__ pooled,
    const float* __restrict__ w1, const float* __restrict__ b1,
    const float* __restrict__ w2, const float* __restrict__ b2,
    const float* __restrict__ w3, const float* __restrict__ b3,
    const float* __restrict__ w4, const float* __restrict__ b4,
    float* __restrict__ out)
{
    __shared__ float p[128], o1[64], o2[32], o3[32];
    const int g = blockIdx.x, t = threadIdx.x;
    p[t] = pooled[g * 128 + t];
    p[t + 64] = pooled[g * 128 + t + 64];
    __syncthreads();
    {
        float s = b1[t];
        for (int k = 0; k < 128; ++k) s += p[k] * w1[k * 64 + t];
        o1[t] = elu1(s);
    }
    __syncthreads();
    if (t < 32) {
        float s = b2[t];
        for (int k = 0; k < 64; ++k) s += o1[k] * w2[k * 32 + t];
        o2[t] = elu1(s);
    }
    __syncthreads();
    if (t < 32) {
        float s = b3[t];
        for (int k = 0; k < 32; ++k) s += o2[k] * w3[k * 32 + t];
        o3[t] = elu1(s);
    }
    __syncthreads();
    if (t < 6) {
        float s = b4[t];
        for (int k = 0; k < 32; ++k) s += o3[k] * w4[k * 6 + t];
        out[g * 6 + t] = s;
    }
}

// batch_pf passthrough: value = graph index, written as float.
__global__ __launch_bounds__(256) void k_batch(float* __restrict__ out)
{
    const int i = blockIdx.x * blockDim.x + threadIdx.x;
    if (i < 16384) out[i] = (float)(i >> 7);
}

// ---------------------------------------------------------------------------
extern "C" void kernel_launch(void* const* d_in, const int* in_sizes, int n_in,
                              void* d_out, int out_size, void* d_ws, size_t ws_size,
                              hipStream_t stream)
{
    (void)in_sizes; (void)n_in; (void)out_size; (void)ws_size;
    const float* x   = (const float*)d_in[0];
    const float* ew1 = (const float*)d_in[1];
    const float* eb1 = (const float*)d_in[2];
    const float* ew2 = (const float*)d_in[3];
    const float* eb2 = (const float*)d_in[4];
    const float* cw[3] = {(const float*)d_in[5], (const float*)d_in[7], (const float*)d_in[9]};
    const float* cb[3] = {(const float*)d_in[6], (const float*)d_in[8], (const float*)d_in[10]};
    const float* ow1 = (const float*)d_in[11];
    const float* ob1 = (const float*)d_in[12];
    const float* ow2 = (const float*)d_in[13];
    const float* ob2 = (const float*)d_in[14];
    const float* ow3 = (const float*)d_in[15];
    const float* ob3 = (const float*)d_in[16];
    const float* ow4 = (const float*)d_in[17];
    const float* ob4 = (const float*)d_in[18];

    float* out = (float*)d_out;
    char* ws = (char*)d_ws;
    float* h      = (float*)(ws);                          // 8 MB
    float* A      = (float*)(ws + (size_t)(8u  << 20));    // 8 MB
    float* Bm     = (float*)(ws + (size_t)(16u << 20));    // 8 MB
    int*   idx    = (int*)  (ws + (size_t)(24u << 20));    // 1.5 MB
    float* pooled = (float*)(ws + (size_t)(26u << 20));    // 64 KB

    k_encoder<<<128, 256, 0, stream>>>(x, ew1, eb1, ew2, eb2, h);
    for (int l = 0; l < 3; ++l) {
        k_knn   <<<128, 256, 0, stream>>>(h, idx);
        k_convAB<<<128, 256, 0, stream>>>(h, cw[l], cb[l], A, Bm);
        k_edge  <<<128, 256, 0, stream>>>(A, Bm, idx, h);
    }
    k_pool <<<128, 128, 0, stream>>>(h, pooled);
    k_mlp  <<<128, 64, 0, stream>>>(pooled, ow1, ob1, ow2, ob2, ow3, ob3, ow4, ob4, out);
    k_batch<<<64, 256, 0, stream>>>(out + 768);
}